// GraphAttentionLayer_40853728919688
// MI455X (gfx1250) — compile-verified
//
#include <hip/hip_runtime.h>
#include <math.h>

typedef __attribute__((ext_vector_type(2))) float v2f;
typedef __attribute__((ext_vector_type(8))) float v8f;

#define IN_DIM        128
#define OUT_DIM       64
#define TARGET_LEN    8192
#define NEIGHBOR_LEN  16384
#define DEG           32
#define E_TOTAL       (TARGET_LEN * DEG)
#define ALPHA         0.2f

__device__ __forceinline__ float leaky(float x) { return (x >= 0.f) ? x : ALPHA * x; }
__device__ __forceinline__ float elu(float x)   { return (x > 0.f) ? x : expm1f(x); }

// ---------------------------------------------------------------------------
// Kernel 1: H = features @ W   (16384x128 @ 128x64), f32 WMMA 16x16x4.
// One wave owns a 16-row x 64-col tile: 4 accumulators, K-loop of 32 steps.
// A frag (16x4 f32, ISA 7.12.2): lanes 0-15 hold K={k0,k0+1}, lanes 16-31 K={k0+2,k0+3}.
// B frag (4x16 f32) mirrors: row-striped across lanes, same K-halving.
// D (16x16 f32): VGPR v -> row v (lanes 0-15) / v+8 (lanes 16-31), col = lane%16.
// ---------------------------------------------------------------------------
__global__ __launch_bounds__(128) void gat_gemm_h(
    const float* __restrict__ X,   // [NEIGHBOR_LEN, IN_DIM]
    const float* __restrict__ W,   // [IN_DIM, OUT_DIM]
    float* __restrict__ H)         // [NEIGHBOR_LEN, OUT_DIM]
{
    const int lane  = threadIdx.x & 31;
    const int wave  = threadIdx.x >> 5;
    const int rbase = (blockIdx.x * 4 + wave) * 16;
    const int mrow  = rbase + (lane & 15);
    const int khalf = (lane < 16) ? 0 : 2;
    const int ncol  = lane & 15;

    v8f acc0 = {}, acc1 = {}, acc2 = {}, acc3 = {};
    for (int k0 = 0; k0 < IN_DIM; k0 += 4) {
        v2f a = *(const v2f*)(X + mrow * IN_DIM + k0 + khalf);
        const float* wr0 = W + (k0 + khalf) * OUT_DIM;
        const float* wr1 = wr0 + OUT_DIM;
        v2f b0 = { wr0[ncol +  0], wr1[ncol +  0] };
        v2f b1 = { wr0[ncol + 16], wr1[ncol + 16] };
        v2f b2 = { wr0[ncol + 32], wr1[ncol + 32] };
        v2f b3 = { wr0[ncol + 48], wr1[ncol + 48] };
        acc0 = __builtin_amdgcn_wmma_f32_16x16x4_f32(false, a, false, b0, (short)0, acc0, false, false);
        acc1 = __builtin_amdgcn_wmma_f32_16x16x4_f32(false, a, false, b1, (short)0, acc1, false, false);
        acc2 = __builtin_amdgcn_wmma_f32_16x16x4_f32(false, a, false, b2, (short)0, acc2, false, false);
        acc3 = __builtin_amdgcn_wmma_f32_16x16x4_f32(false, a, false, b3, (short)0, acc3, false, false);
    }
    const int roff = (lane >= 16) ? 8 : 0;
#pragma unroll
    for (int v = 0; v < 8; ++v) {
        float* o = H + (rbase + roff + v) * OUT_DIM + ncol;
        o[0]  = acc0[v];
        o[16] = acc1[v];
        o[32] = acc2[v];
        o[48] = acc3[v];
    }
}

// ---------------------------------------------------------------------------
// Kernel 2: NLH = nl_features[nidx] @ W  (gathered rows) + elu(NLH) -> OUT2.
// ---------------------------------------------------------------------------
__global__ __launch_bounds__(128) void gat_gemm_nlh(
    const float* __restrict__ X,      // nl_features [NEIGHBOR_LEN, IN_DIM]
    const float* __restrict__ W,
    const int*   __restrict__ nidx,   // [TARGET_LEN]
    float* __restrict__ NLH,          // [TARGET_LEN, OUT_DIM] raw
    float* __restrict__ OUT2)         // elu(NLH)
{
    const int lane  = threadIdx.x & 31;
    const int wave  = threadIdx.x >> 5;
    const int rbase = (blockIdx.x * 4 + wave) * 16;
    const int srow  = nidx[rbase + (lane & 15)];
    const int khalf = (lane < 16) ? 0 : 2;
    const int ncol  = lane & 15;

    v8f acc0 = {}, acc1 = {}, acc2 = {}, acc3 = {};
    for (int k0 = 0; k0 < IN_DIM; k0 += 4) {
        v2f a = *(const v2f*)(X + srow * IN_DIM + k0 + khalf);
        const float* wr0 = W + (k0 + khalf) * OUT_DIM;
        const float* wr1 = wr0 + OUT_DIM;
        v2f b0 = { wr0[ncol +  0], wr1[ncol +  0] };
        v2f b1 = { wr0[ncol + 16], wr1[ncol + 16] };
        v2f b2 = { wr0[ncol + 32], wr1[ncol + 32] };
        v2f b3 = { wr0[ncol + 48], wr1[ncol + 48] };
        acc0 = __builtin_amdgcn_wmma_f32_16x16x4_f32(false, a, false, b0, (short)0, acc0, false, false);
        acc1 = __builtin_amdgcn_wmma_f32_16x16x4_f32(false, a, false, b1, (short)0, acc1, false, false);
        acc2 = __builtin_amdgcn_wmma_f32_16x16x4_f32(false, a, false, b2, (short)0, acc2, false, false);
        acc3 = __builtin_amdgcn_wmma_f32_16x16x4_f32(false, a, false, b3, (short)0, acc3, false, false);
    }
    const int roff = (lane >= 16) ? 8 : 0;
#pragma unroll
    for (int v = 0; v < 8; ++v) {
        const int row = rbase + roff + v;
        float* o  = NLH  + row * OUT_DIM + ncol;
        float* o2 = OUT2 + row * OUT_DIM + ncol;
        o[0]  = acc0[v];  o2[0]  = elu(acc0[v]);
        o[16] = acc1[v];  o2[16] = elu(acc1[v]);
        o[32] = acc2[v];  o2[32] = elu(acc2[v]);
        o[48] = acc3[v];  o2[48] = elu(acc3[v]);
    }
}

// ---------------------------------------------------------------------------
// Kernel 3: per-row projections s1 = h.a1, s2 = h.a2
// ---------------------------------------------------------------------------
__global__ __launch_bounds__(256) void gat_s12(
    const float* __restrict__ H, const float* __restrict__ a,
    float* __restrict__ S1, float* __restrict__ S2)
{
    const int r = blockIdx.x * blockDim.x + threadIdx.x;
    if (r >= NEIGHBOR_LEN) return;
    const float* hr = H + r * OUT_DIM;
    float s1 = 0.f, s2 = 0.f;
#pragma unroll
    for (int n = 0; n < OUT_DIM; ++n) {
        const float hv = hr[n];
        s1 += hv * a[n];
        s2 += hv * a[OUT_DIM + n];
    }
    S1[r] = s1;
    S2[r] = s2;
}

// ---------------------------------------------------------------------------
// Kernel 4: nl_e = leaky(s1[nidx[r]] + nl_h[r].a2)
// ---------------------------------------------------------------------------
__global__ __launch_bounds__(256) void gat_nle(
    const float* __restrict__ NLH, const float* __restrict__ a,
    const float* __restrict__ S1, const int* __restrict__ nidx,
    float* __restrict__ NLE)
{
    const int r = blockIdx.x * blockDim.x + threadIdx.x;
    if (r >= TARGET_LEN) return;
    const float* hr = NLH + r * OUT_DIM;
    float d = 0.f;
#pragma unroll
    for (int n = 0; n < OUT_DIM; ++n) d += hr[n] * a[OUT_DIM + n];
    NLE[r] = leaky(S1[nidx[r]] + d);
}

// ---------------------------------------------------------------------------
// Kernel 5: per-target-row softmax over {32 edge scores, nl_e} + weighted
// aggregation. DEG == 32 == wave32: one wave per row for the score phase,
// 64 threads per row (coalesced over columns) for the gather-sum phase.
// All NEG_FILL softmax entries underflow to exactly 0, so summing only the
// 33 live slots is exact. Edge columns within a row are distinct, so the
// scatter-then-matmul equals a direct sum over the edge list.
// ---------------------------------------------------------------------------
__global__ __launch_bounds__(128) void gat_agg(
    const float* __restrict__ H, const float* __restrict__ NLH,
    const float* __restrict__ S1, const float* __restrict__ S2,
    const float* __restrict__ NLE, const int* __restrict__ adj,
    float* __restrict__ OUT1)
{
    __shared__ float sw[2][DEG];
    __shared__ int   sc[2][DEG];
    __shared__ float swnl[2];

    const int lane  = threadIdx.x & 31;
    const int wave  = threadIdx.x >> 5;
    const int rbase = blockIdx.x * 2;

    if (wave < 2) {
        const int r  = rbase + wave;
        const int e  = r * DEG + lane;
        const int c0 = adj[e];
        const int c1 = adj[E_TOTAL + e];
        const float ek  = leaky(S1[c0] + S2[c1]);
        const float nle = NLE[r];

        float m = ek;
#pragma unroll
        for (int off = 16; off > 0; off >>= 1) m = fmaxf(m, __shfl_xor(m, off, 32));
        m = fmaxf(m, nle);

        const float w = expf(ek - m);
        float s = w;
#pragma unroll
        for (int off = 16; off > 0; off >>= 1) s += __shfl_xor(s, off, 32);
        const float wnl  = expf(nle - m);
        const float invZ = 1.f / (s + wnl);

        sw[wave][lane] = w * invZ;
        sc[wave][lane] = c1;
        if (lane == 0) swnl[wave] = wnl * invZ;
    }
    __syncthreads();

    const int lrow = threadIdx.x >> 6;   // 0 or 1
    const int n    = threadIdx.x & 63;
    const int r    = rbase + lrow;

    float acc = swnl[lrow] * NLH[r * OUT_DIM + n];
#pragma unroll
    for (int k = 0; k < DEG; ++k)
        acc += sw[lrow][k] * H[sc[lrow][k] * OUT_DIM + n];

    OUT1[r * OUT_DIM + n] = elu(acc);
}

// ---------------------------------------------------------------------------
extern "C" void kernel_launch(void* const* d_in, const int* in_sizes, int n_in,
                              void* d_out, int out_size, void* d_ws, size_t ws_size,
                              hipStream_t stream) {
    const float* features    = (const float*)d_in[0];
    const float* nl_features = (const float*)d_in[1];
    const float* W           = (const float*)d_in[2];
    const float* a           = (const float*)d_in[3];   // [2*OUT_DIM]
    const int*   adj         = (const int*)d_in[4];     // [2, E]
    // d_in[5] = target_index_out (implied by edge ordering, unused)
    const int*   nidx        = (const int*)d_in[6];     // [TARGET_LEN]

    float* OUT1 = (float*)d_out;                        // elu(h_prime) [8192,64]
    float* OUT2 = OUT1 + TARGET_LEN * OUT_DIM;          // elu(nl_h)    [8192,64]

    float* ws  = (float*)d_ws;
    float* H   = ws;                                    // 16384*64
    float* NLH = H   + NEIGHBOR_LEN * OUT_DIM;          //  8192*64
    float* S1  = NLH + TARGET_LEN * OUT_DIM;            // 16384
    float* S2  = S1  + NEIGHBOR_LEN;                    // 16384
    float* NLE = S2  + NEIGHBOR_LEN;                    //  8192

    // 1) H = features @ W         (1024 wave-tiles of 16 rows, 4 waves/block)
    gat_gemm_h<<<NEIGHBOR_LEN / 64, 128, 0, stream>>>(features, W, H);
    // 2) NLH = nl_features[nidx] @ W  (+ elu -> OUT2)
    gat_gemm_nlh<<<TARGET_LEN / 64, 128, 0, stream>>>(nl_features, W, nidx, NLH, OUT2);
    // 3) s1, s2
    gat_s12<<<NEIGHBOR_LEN / 256, 256, 0, stream>>>(H, a, S1, S2);
    // 4) nl_e
    gat_nle<<<TARGET_LEN / 256, 256, 0, stream>>>(NLH, a, S1, nidx, NLE);
    // 5) softmax + aggregate + elu -> OUT1   (2 rows per 128-thread block)
    gat_agg<<<TARGET_LEN / 2, 128, 0, stream>>>(H, NLH, S1, S2, NLE, adj, OUT1);
}